// WavAddRIR_3745211482893
// MI455X (gfx1250) — compile-verified
//
#include <hip/hip_runtime.h>
#include <hip/hip_bf16.h>

// ---------------------------------------------------------------------------
// WavAddRIR on MI455X (gfx1250): four-step FFT convolution.
//   nfft = 2^19 = N1(1024) x N2(512)
//   forward  = rowMajor: colFFT(1024, stride 512) -> 4-step twiddle -> rowFFT(512)
//   inverse  = rowFFT(conj) -> conj twiddle -> colFFT(conj), scaled 1/N at pointwise
// Radix-16 butterflies run as 16x16x16 real GEMMs on V_WMMA_F32_16X16X4_F32
// (4 chained K=4 ops per GEMM, 16 WMMAs per complex DFT16).
// Tile fills use GLOBAL_LOAD_ASYNC_TO_LDS_B64/B128 (ASYNCcnt) so global->LDS
// traffic bypasses the VGPR file while the matrix pipes work.
// Workspace layout (assumed ws_size >= ~139 MB):
//   [0)                 : 33 * 2^19 complex f32 (audio rows 0..31, RIR row 32)
//   [specBytes)         : origAmp[32], outAmp[32], delay(int)
// ---------------------------------------------------------------------------

typedef float v2f __attribute__((ext_vector_type(2)));
typedef float v8f __attribute__((ext_vector_type(8)));

#define NFFT   (1 << 19)
#define N1     1024
#define N2     512
#define BROWS  32
#define TLEN   480000
#define KLEN   16000
#define NIMG   33          // 32 audio rows + 1 RIR row
#define LDP    17          // LDS row pitch (complex) to avoid bank conflicts
#define TWO_PI 6.28318530717958647692f

// LDS byte offset of a generic pointer into shared memory (ptrtoint of AS(3)).
__device__ inline unsigned lds_offset(const void* p) {
  return (unsigned)(unsigned long long)(__attribute__((address_space(3))) const char*)p;
}

// ----------------------------- WMMA DFT-16 helpers -------------------------

// A-matrix (16x4 f32) layout: lanes 0-15 hold rows M=0..15; VGPR0 = {K, K+2},
// VGPR1 = {K+1, K+3} split across lane halves. Build the DFT16 matrix chunks.
__device__ inline void make_w16(float sgn, v2f wr[4], v2f wi[4]) {
  const int lane = threadIdx.x & 31;
  const int r    = lane & 15;
  const int kh   = (lane >> 4) << 1;   // 0 or 2
#pragma unroll
  for (int i = 0; i < 4; ++i) {
    const int k0 = 4 * i + kh;
    const float a0 = sgn * TWO_PI * (float)((r * k0) & 15) * (1.0f / 16.0f);
    const float a1 = sgn * TWO_PI * (float)((r * (k0 + 1)) & 15) * (1.0f / 16.0f);
    wr[i].x = __cosf(a0); wi[i].x = __sinf(a0);
    wr[i].y = __cosf(a1); wi[i].y = __sinf(a1);
  }
}

// Complex 16-point DFT on a batch of 16 columns held in LDS:
//   Y[r][col] = sum_m W16[r][m] * X[base + stride*m][col]
// B-matrix (4x16 f32): lanes hold N=0..15 (col); VGPR0 = rows {K, K+2},
// VGPR1 = rows {K+1, K+3} split across lane halves (mirrors A layout).
__device__ inline void dft16_gemm(const v2f wr[4], const v2f wi[4],
                                  const float2* tile, int base, int stride,
                                  v8f& accr, v8f& acci) {
  const int lane = threadIdx.x & 31;
  const int col  = lane & 15;
  const int kh   = (lane >> 4) << 1;
#pragma unroll
  for (int v = 0; v < 8; ++v) { accr[v] = 0.0f; acci[v] = 0.0f; }
#pragma unroll
  for (int i = 0; i < 4; ++i) {
    const int m0 = 4 * i + kh;
    const float2 x0 = tile[(base + stride * m0)       * LDP + col];
    const float2 x1 = tile[(base + stride * (m0 + 1)) * LDP + col];
    v2f br; br.x = x0.x; br.y = x1.x;
    v2f bi; bi.x = x0.y; bi.y = x1.y;
    const v2f wni = -wi[i];   // f32 WMMA has no neg_a modifier; negate operand
    // Yr += Wr*Xr - Wi*Xi
    accr = __builtin_amdgcn_wmma_f32_16x16x4_f32(false, wr[i], false, br,
                                                 (short)0, accr, false, false);
    accr = __builtin_amdgcn_wmma_f32_16x16x4_f32(false, wni,   false, bi,
                                                 (short)0, accr, false, false);
    // Yi += Wr*Xi + Wi*Xr
    acci = __builtin_amdgcn_wmma_f32_16x16x4_f32(false, wr[i], false, bi,
                                                 (short)0, acci, false, false);
    acci = __builtin_amdgcn_wmma_f32_16x16x4_f32(false, wi[i], false, br,
                                                 (short)0, acci, false, false);
  }
}

// D-matrix layout: VGPR v, lanes 0-15 -> row v, lanes 16-31 -> row v+8.
// Apply DIF stage twiddle W_L^(j*r) and write back in place.
__device__ inline void dft16_store(const v8f& ar, const v8f& ai, float2* tile,
                                   int base, int stride, int j, int L, float sgn) {
  const int lane = threadIdx.x & 31;
  const int col  = lane & 15;
  const int rhi  = (lane >> 4) << 3;
  const float ang0 = sgn * TWO_PI * (float)j / (float)L;
#pragma unroll
  for (int v = 0; v < 8; ++v) {
    const int r = v + rhi;
    float sn, cs;
    __sincosf(ang0 * (float)r, &sn, &cs);
    const float xr = ar[v], xi = ai[v];
    tile[(base + stride * r) * LDP + col] =
        make_float2(xr * cs - xi * sn, xr * sn + xi * cs);
  }
}

__device__ inline int rev1024(int p) {  // DIF radices [16,16,4] -> natural order
  return (p >> 6) | (((p >> 2) & 15) << 4) | ((p & 3) << 8);
}
__device__ inline int rev512(int p) {   // DIF radices [16,16,2]
  return (p >> 5) | (((p >> 1) & 15) << 4) | ((p & 1) << 8);
}

// ----------------------------- FFT kernels ---------------------------------

// 1024-point FFTs down columns (stride N2). One block = 16 adjacent columns.
// dir = -1: FFT, then apply four-step twiddle at store.
// dir = +1: apply conjugate four-step twiddle, then inverse FFT (unscaled).
__global__ __launch_bounds__(256)
void fft_col_kernel(float2* __restrict__ spec, int dir) {
  extern __shared__ float2 tile[];              // N1 * LDP complex
  const int tilesPerImg = N2 / 16;              // 32
  const int img     = blockIdx.x / tilesPerImg;
  const int colBase = (blockIdx.x % tilesPerImg) * 16;
  const float sgn   = (float)dir;
  float2* __restrict__ base = spec + (size_t)img * NFFT;
  const int tid = threadIdx.x;

  // async global->LDS tile fill: 16B granules (2 complex), 128B per tile row
  {
    const unsigned ldsBase = lds_offset(tile);
    for (int g = tid; g < N1 * 8; g += blockDim.x) {
      const int r = g >> 3, c8 = (g & 7) << 1;
      const unsigned long long ga =
          (unsigned long long)(const void*)(base + (size_t)r * N2 + (colBase + c8));
      const unsigned la = ldsBase + (unsigned)((r * LDP + c8) * sizeof(float2));
      asm volatile("global_load_async_to_lds_b128 %0, %1, off"
                   :: "v"(la), "v"(ga) : "memory");
    }
    asm volatile("s_wait_asynccnt 0x0" ::: "memory");
  }
  __syncthreads();

  if (dir > 0) {  // inverse: undo the four-step twiddle first
    for (int idx = tid; idx < N1 * 16; idx += blockDim.x) {
      const int r = idx >> 4, c = idx & 15;
      const float ang = sgn * TWO_PI * ((float)r * (float)(colBase + c)) / (float)NFFT;
      float sn, cs; __sincosf(ang, &sn, &cs);
      const float2 x = tile[r * LDP + c];
      tile[r * LDP + c] = make_float2(x.x * cs - x.y * sn, x.x * sn + x.y * cs);
    }
    __syncthreads();
  }

  v2f wr[4], wi[4];
  make_w16(sgn, wr, wi);
  const int wave = tid >> 5, nw = (int)blockDim.x >> 5;

  // stage 1: L=1024, stride=64 (64 WMMA tasks)
  for (int t = wave; t < 64; t += nw) {
    v8f ar, ai;
    dft16_gemm(wr, wi, tile, t, 64, ar, ai);
    dft16_store(ar, ai, tile, t, 64, t, 1024, sgn);
  }
  __syncthreads();
  // stage 2: L=64, stride=4 (64 WMMA tasks)
  for (int t = wave; t < 64; t += nw) {
    const int blk = t >> 2, j = t & 3, b0 = blk * 64 + j;
    v8f ar, ai;
    dft16_gemm(wr, wi, tile, b0, 4, ar, ai);
    dft16_store(ar, ai, tile, b0, 4, j, 64, sgn);
  }
  __syncthreads();
  // stage 3: radix-4, stride 1 (VALU)
  for (int g = tid; g < (N1 / 4) * 16; g += blockDim.x) {
    const int c = g & 15, b0 = (g >> 4) * 4;
    const float2 x0 = tile[(b0 + 0) * LDP + c];
    const float2 x1 = tile[(b0 + 1) * LDP + c];
    const float2 x2 = tile[(b0 + 2) * LDP + c];
    const float2 x3 = tile[(b0 + 3) * LDP + c];
    const float t0r = x0.x + x2.x, t0i = x0.y + x2.y;
    const float t1r = x0.x - x2.x, t1i = x0.y - x2.y;
    const float t2r = x1.x + x3.x, t2i = x1.y + x3.y;
    const float t3r = x1.x - x3.x, t3i = x1.y - x3.y;
    tile[(b0 + 0) * LDP + c] = make_float2(t0r + t2r, t0i + t2i);
    tile[(b0 + 2) * LDP + c] = make_float2(t0r - t2r, t0i - t2i);
    tile[(b0 + 1) * LDP + c] = make_float2(t1r - sgn * t3i, t1i + sgn * t3r);
    tile[(b0 + 3) * LDP + c] = make_float2(t1r + sgn * t3i, t1i - sgn * t3r);
  }
  __syncthreads();

  // store digit-reversed -> natural order; forward applies four-step twiddle
  for (int idx = tid; idx < N1 * 16; idx += blockDim.x) {
    const int p = idx >> 4, c = idx & 15;
    const int k1 = rev1024(p);
    float2 x = tile[p * LDP + c];
    if (dir < 0) {
      const float ang = sgn * TWO_PI * ((float)k1 * (float)(colBase + c)) / (float)NFFT;
      float sn, cs; __sincosf(ang, &sn, &cs);
      x = make_float2(x.x * cs - x.y * sn, x.x * sn + x.y * cs);
    }
    base[(size_t)k1 * N2 + (colBase + c)] = x;
  }
}

// 512-point FFTs along contiguous rows. One block = 16 adjacent rows.
__global__ __launch_bounds__(256)
void fft_row_kernel(float2* __restrict__ spec, int dir) {
  extern __shared__ float2 tile[];              // N2 * LDP complex
  const int tilesPerImg = N1 / 16;              // 64
  const int img     = blockIdx.x / tilesPerImg;
  const int rowBase = (blockIdx.x % tilesPerImg) * 16;
  const float sgn   = (float)dir;
  float2* __restrict__ base = spec + (size_t)img * NFFT + (size_t)rowBase * N2;
  const int tid = threadIdx.x;

  // async global->LDS fill with transpose: coalesced 8B reads, scattered LDS
  {
    const unsigned ldsBase = lds_offset(tile);
    for (int g = tid; g < N2 * 16; g += blockDim.x) {
      const int rl = g >> 9, p = g & (N2 - 1);
      const unsigned long long ga =
          (unsigned long long)(const void*)(base + (size_t)rl * N2 + p);
      const unsigned la = ldsBase + (unsigned)((p * LDP + rl) * sizeof(float2));
      asm volatile("global_load_async_to_lds_b64 %0, %1, off"
                   :: "v"(la), "v"(ga) : "memory");
    }
    asm volatile("s_wait_asynccnt 0x0" ::: "memory");
  }
  __syncthreads();

  v2f wr[4], wi[4];
  make_w16(sgn, wr, wi);
  const int wave = tid >> 5, nw = (int)blockDim.x >> 5;

  // stage 1: L=512, stride=32 (32 WMMA tasks)
  for (int t = wave; t < 32; t += nw) {
    v8f ar, ai;
    dft16_gemm(wr, wi, tile, t, 32, ar, ai);
    dft16_store(ar, ai, tile, t, 32, t, 512, sgn);
  }
  __syncthreads();
  // stage 2: L=32, stride=2 (32 WMMA tasks)
  for (int t = wave; t < 32; t += nw) {
    const int blk = t >> 1, j = t & 1, b0 = blk * 32 + j;
    v8f ar, ai;
    dft16_gemm(wr, wi, tile, b0, 2, ar, ai);
    dft16_store(ar, ai, tile, b0, 2, j, 32, sgn);
  }
  __syncthreads();
  // stage 3: radix-2
  for (int g = tid; g < (N2 / 2) * 16; g += blockDim.x) {
    const int c = g & 15, b0 = (g >> 4) * 2;
    const float2 x0 = tile[(b0 + 0) * LDP + c];
    const float2 x1 = tile[(b0 + 1) * LDP + c];
    tile[(b0 + 0) * LDP + c] = make_float2(x0.x + x1.x, x0.y + x1.y);
    tile[(b0 + 1) * LDP + c] = make_float2(x0.x - x1.x, x0.y - x1.y);
  }
  __syncthreads();

  for (int idx = tid; idx < N2 * 16; idx += blockDim.x) {
    const int rl = idx >> 9, p = idx & (N2 - 1);
    base[(size_t)rl * N2 + p] = tile[rev512(p) * LDP + rl];  // coalesced write
  }
}

// ----------------------------- glue kernels --------------------------------

__global__ __launch_bounds__(256)
void prep_pad_kernel(const float* __restrict__ audio, const float* __restrict__ rir,
                     float2* __restrict__ spec) {
  const size_t i = (size_t)blockIdx.x * blockDim.x + threadIdx.x;  // < NIMG*NFFT
  const int img = (int)(i >> 19);
  const int t   = (int)(i & (NFFT - 1));
  float re = 0.0f;
  if (img < BROWS) {
    if (t < TLEN) re = audio[(size_t)img * TLEN + t];
  } else {
    if (t < KLEN) re = rir[t];
  }
  spec[i] = make_float2(re, 0.0f);
}

__global__ __launch_bounds__(256)
void row_absmean_kernel(const float* __restrict__ x, float* __restrict__ out) {
  __shared__ float sdata[256];
  const int b = blockIdx.x;
  float acc = 0.0f;
  for (int t = threadIdx.x; t < TLEN; t += blockDim.x)
    acc += fabsf(x[(size_t)b * TLEN + t]);
  sdata[threadIdx.x] = acc;
  __syncthreads();
  for (int s = 128; s > 0; s >>= 1) {
    if ((int)threadIdx.x < s) sdata[threadIdx.x] += sdata[threadIdx.x + s];
    __syncthreads();
  }
  if (threadIdx.x == 0) out[b] = sdata[0] / (float)TLEN;
}

__global__ __launch_bounds__(256)
void rir_argmax_kernel(const float* __restrict__ rir, int* __restrict__ delayOut) {
  __shared__ float sv[256];
  __shared__ int   si[256];
  const int tid = threadIdx.x;
  float bv = -1.0f; int bi = 0;
  for (int t = tid; t < KLEN; t += blockDim.x) {
    const float v = fabsf(rir[t]);
    if (v > bv) { bv = v; bi = t; }
  }
  sv[tid] = bv; si[tid] = bi;
  __syncthreads();
  for (int s = 128; s > 0; s >>= 1) {
    if (tid < s) {
      if (sv[tid + s] > sv[tid] ||
          (sv[tid + s] == sv[tid] && si[tid + s] < si[tid])) {
        sv[tid] = sv[tid + s]; si[tid] = si[tid + s];
      }
    }
    __syncthreads();
  }
  if (tid == 0) *delayOut = si[0];
}

__global__ __launch_bounds__(256)
void pointwise_mul_kernel(float2* __restrict__ spec) {
  const size_t i = (size_t)blockIdx.x * blockDim.x + threadIdx.x;  // < BROWS*NFFT
  const size_t k = i & (NFFT - 1);
  const float2 a = spec[i];
  const float2 r = spec[(size_t)BROWS * NFFT + k];
  const float scale = 1.0f / (float)NFFT;  // fold inverse-FFT 1/N here
  spec[i] = make_float2((a.x * r.x - a.y * r.y) * scale,
                        (a.x * r.y + a.y * r.x) * scale);
}

__global__ __launch_bounds__(256)
void conv_absmean_kernel(const float2* __restrict__ spec,
                         const int* __restrict__ delay,
                         float* __restrict__ out) {
  __shared__ float sdata[256];
  const int b = blockIdx.x;
  const int d = *delay;
  float acc = 0.0f;
  for (int t = threadIdx.x; t < TLEN; t += blockDim.x)
    acc += fabsf(spec[(size_t)b * NFFT + t + d].x);
  sdata[threadIdx.x] = acc;
  __syncthreads();
  for (int s = 128; s > 0; s >>= 1) {
    if ((int)threadIdx.x < s) sdata[threadIdx.x] += sdata[threadIdx.x + s];
    __syncthreads();
  }
  if (threadIdx.x == 0) out[b] = sdata[0] / (float)TLEN;
}

__global__ __launch_bounds__(256)
void post_write_kernel(const float2* __restrict__ spec,
                       const int* __restrict__ delay,
                       const float* __restrict__ origAmp,
                       const float* __restrict__ outAmp,
                       float* __restrict__ out) {
  const size_t i = (size_t)blockIdx.x * blockDim.x + threadIdx.x;  // < BROWS*TLEN
  const int b = (int)(i / TLEN);
  const int t = (int)(i % TLEN);
  const int d = *delay;
  const float v = spec[(size_t)b * NFFT + t + d].x;
  out[i] = v * origAmp[b] / (outAmp[b] + 1e-14f);
}

// ----------------------------- launch --------------------------------------

extern "C" void kernel_launch(void* const* d_in, const int* in_sizes, int n_in,
                              void* d_out, int out_size, void* d_ws, size_t ws_size,
                              hipStream_t stream) {
  (void)in_sizes; (void)n_in; (void)out_size; (void)ws_size;
  const float* audio = (const float*)d_in[0];
  const float* rir   = (const float*)d_in[1];
  float*       out   = (float*)d_out;

  float2* spec = (float2*)d_ws;
  const size_t specElems = (size_t)NIMG * NFFT;
  float* aux     = (float*)((char*)d_ws + specElems * sizeof(float2));
  float* origAmp = aux;            // 32 floats
  float* outAmp  = aux + 32;       // 32 floats
  int*   delay   = (int*)(aux + 64);

  const int colLds = N1 * LDP * (int)sizeof(float2);  // 139264 B
  const int rowLds = N2 * LDP * (int)sizeof(float2);  //  69632 B

  // 1) zero-pad into complex workspace (33 rows)
  prep_pad_kernel<<<(int)(specElems / 256), 256, 0, stream>>>(audio, rir, spec);
  // 2) per-row original amplitude; RIR direct-path delay
  row_absmean_kernel<<<BROWS, 256, 0, stream>>>(audio, origAmp);
  rir_argmax_kernel<<<1, 256, 0, stream>>>(rir, delay);
  // 3) forward four-step FFT on all 33 rows
  fft_col_kernel<<<NIMG * (N2 / 16), 256, colLds, stream>>>(spec, -1);
  fft_row_kernel<<<NIMG * (N1 / 16), 256, rowLds, stream>>>(spec, -1);
  // 4) spectrum multiply (audio rows x RIR row), 1/N folded in
  pointwise_mul_kernel<<<(int)((size_t)BROWS * NFFT / 256), 256, 0, stream>>>(spec);
  // 5) inverse FFT on the 32 audio rows (reverse order, conjugate twiddles)
  fft_row_kernel<<<BROWS * (N1 / 16), 256, rowLds, stream>>>(spec, +1);
  fft_col_kernel<<<BROWS * (N2 / 16), 256, colLds, stream>>>(spec, +1);
  // 6) output amplitude, then rescale + delay-aligned slice
  conv_absmean_kernel<<<BROWS, 256, 0, stream>>>(spec, delay, outAmp);
  post_write_kernel<<<(int)((size_t)BROWS * TLEN / 256), 256, 0, stream>>>(
      spec, delay, origAmp, outAmp, out);
}